// HybridDistanceModule_15522011807807
// MI455X (gfx1250) — compile-verified
//
#include <hip/hip_runtime.h>
#include <hip/hip_bf16.h>

typedef unsigned int u32;
typedef unsigned long long u64;
typedef __attribute__((ext_vector_type(4))) u32 u32x4;
typedef __attribute__((ext_vector_type(8))) u32 u32x8;

#define TILE_N 64
#define DIM 64
#define NQ 64
#define LDS_ROW (DIM + 1)   // TDM pads 1 dword per 64 -> stride 65 dwords, conflict-free
#define CH 16               // sort: elements fused per thread-chunk (16 x u64 = 32 VGPRs)
#define LN2 0.6931471805599453f

// Single dynamic-LDS symbol shared by both kernels (cast per kernel).
extern __shared__ __align__(16) unsigned char dyn_smem[];

// ---------------------------------------------------------------------------
// Tensor DMA descriptor (D#) for a 2D f32 tile load: tensor is [n_total x DIM]
// row-major f32; tile is [tile_rows x DIM]; destination LDS gets 1 dword of
// padding after every 64 dwords (CDNA5 ISA) -> row stride 65.
// ---------------------------------------------------------------------------
struct TdmDesc { u32x4 g0; u32x8 g1; };

__device__ __forceinline__ TdmDesc tdm_make_2d(const float* gtile, u32 lds_byte_addr,
                                               int n_total, int tile_rows) {
  TdmDesc d;
  u64 ga = (u64)(size_t)gtile;
  d.g0.x = 1u;                                               // count=1, user mode, no gather
  d.g0.y = lds_byte_addr;                                    // lds_addr (bytes)
  d.g0.z = (u32)(ga & 0xFFFFFFFFull);                        // global_addr[31:0]
  d.g0.w = (u32)((ga >> 32) & 0x1FFFFFFull) | (2u << 30);    // global_addr[56:32] | type=2
  d.g1.s0 = (2u << 16)    // data_size = 4 bytes
          | (1u << 20)    // pad_enable (load only)
          | (5u << 22)    // pad_interval code 5 = every 64 dwords
          | (0u << 25);   // pad_amount code 0 = 1 dword (workgroup_mask=0: not in cluster)
  d.g1.s1 = ((u32)DIM & 0xFFFFu) << 16;                      // tensor_dim0[15:0]
  d.g1.s2 = ((u32)n_total & 0xFFFFu) << 16;                  // dim0[31:16]=0 | tensor_dim1[15:0]
  d.g1.s3 = (((u32)n_total >> 16) & 0xFFFFu)                 // tensor_dim1[31:16]
          | ((u32)DIM << 16);                                // tile_dim0 = 64
  d.g1.s4 = (u32)tile_rows & 0xFFFFu;                        // tile_dim1 | tile_dim2=0
  d.g1.s5 = (u32)DIM;                                        // tensor_dim0_stride[31:0] = 64
  d.g1.s6 = 0u;
  d.g1.s7 = 0u;
  return d;
}

// ---------------------------------------------------------------------------
// Phase 1: neg[q,n] = 0.5 * sum_d [ (qf-gf)^2 / s + log s ],  s = e^{2sq}+e^{2sn}.
// Staged as half-values s2 = s/2 so that the per-d log can be replaced by a
// single log of the running product:
//   neg = 0.25*sum diff^2*rcp(s2) + 0.5*log(prod s2) + 32*ln2
// Gallery tile staged by the Tensor Data Mover into bank-padded LDS; all 64
// query rows (32 KB) staged by plain loads. Thread t: gallery column t&63,
// 16 queries (t>>6)*16..+15 in registers (broadcast LDS reads per wave).
// ---------------------------------------------------------------------------
__global__ void elk_sims_kernel(const float* __restrict__ q_feat,
                                const float* __restrict__ q_sig,
                                const float* __restrict__ g_feat,
                                const float* __restrict__ g_sig,
                                float* __restrict__ neg, int N) {
  float* smem = (float*)dyn_smem;
  float* s_qf = smem;                     // NQ*DIM
  float* s_qe = s_qf + NQ * DIM;          // NQ*DIM: 0.5*exp(2*sigma_q)
  float* s_gf = s_qe + NQ * DIM;          // TILE_N*LDS_ROW (TDM-padded)
  float* s_ge = s_gf + TILE_N * LDS_ROW;  // TILE_N*LDS_ROW: raw -> 0.5*exp(2*sigma_g)

  const int tid = threadIdx.x;
  const int n0  = blockIdx.x * TILE_N;
  int rows = N - n0; if (rows > TILE_N) rows = TILE_N;

  for (int i = tid; i < NQ * DIM; i += 256) {
    s_qf[i] = q_feat[i];
    s_qe[i] = 0.5f * __expf(2.0f * q_sig[i]);
  }

  // Async tensor DMA: gallery feature tile + raw sigma tile into padded LDS.
  TdmDesc df  = tdm_make_2d(g_feat + (size_t)n0 * DIM, (u32)(size_t)s_gf, N, rows);
  TdmDesc dsg = tdm_make_2d(g_sig  + (size_t)n0 * DIM, (u32)(size_t)s_ge, N, rows);
  if (tid < 32) {  // wave 0 only; TDM ignores EXEC, tracked by TENSORcnt
    asm volatile("tensor_load_to_lds %0, %1" :: "s"(df.g0),  "s"(df.g1)  : "memory");
    asm volatile("tensor_load_to_lds %0, %1" :: "s"(dsg.g0), "s"(dsg.g1) : "memory");
    __builtin_amdgcn_s_wait_tensorcnt(0);
  }
  __syncthreads();

  // In-place sigma -> 0.5*exp(2*sigma) on the padded tile (pad slots harmless).
  for (int i = tid; i < TILE_N * LDS_ROW; i += 256)
    s_ge[i] = 0.5f * __expf(2.0f * s_ge[i]);
  __syncthreads();

  const int nl = tid & (TILE_N - 1);   // gallery column in tile
  const int qb = (tid >> 6) << 4;      // 16-query batch
  float accA[16], prod[16];
#pragma unroll
  for (int i = 0; i < 16; ++i) { accA[i] = 0.0f; prod[i] = 1.0f; }

  if (nl < rows) {
#pragma unroll 4
    for (int d = 0; d < DIM; ++d) {
      const float gfv = s_gf[nl * LDS_ROW + d];
      const float gev = s_ge[nl * LDS_ROW + d];
#pragma unroll
      for (int i = 0; i < 16; ++i) {
        const int q = qb + i;
        const float diff = s_qf[q * DIM + d] - gfv;
        const float s2   = s_qe[q * DIM + d] + gev;            // s/2 in [~0.6, 2.7]
        accA[i] = fmaf(diff * diff, __builtin_amdgcn_rcpf(s2), accA[i]);
        prod[i] *= s2;                                          // no overflow in f32
      }
    }
#pragma unroll
    for (int i = 0; i < 16; ++i)
      neg[(size_t)(qb + i) * N + (n0 + nl)] =
          0.25f * accA[i] + 0.5f * __logf(prod[i]) + 32.0f * LN2;
  }
}

// ---------------------------------------------------------------------------
// Phase 2: per-query full sort + argsort. One 1024-thread workgroup per query;
// bitonic sort of NPAD=32768 packed u64 keys entirely in LDS.
// Bank-conflict avoidance: 2-dword pad every 16 keys -> phys = i + (i>>4)
// (272 KB <= 320 KB WGP LDS; 8B alignment preserved; 32 distinct banks/wave
// for both strided and chunked access). Low stages j<=8 are fused in
// registers per aligned 16-element chunk (single LDS load+store per phase).
// Key = order-preserving f32 transform in high word, index low word ->
// ties resolve by ascending index (matches stable argsort).
// ---------------------------------------------------------------------------
__device__ __forceinline__ int kphys(int i) { return i + (i >> 4); }

__global__ void elk_sort_kernel(const float* __restrict__ neg,
                                float* __restrict__ out_vals,
                                int* __restrict__ out_idx,
                                int N, int NPAD) {
  u64* keys = (u64*)dyn_smem;
  const int q   = blockIdx.x;
  const int tid = threadIdx.x;
  const int nt  = blockDim.x;
  const float* row = neg + (size_t)q * N;

  for (int i = tid; i < NPAD; i += nt) {
    u64 p;
    if (i < N) {
      const u32 b   = __float_as_uint(row[i]);
      const u32 key = (b & 0x80000000u) ? ~b : (b | 0x80000000u);
      p = ((u64)key << 32) | (u32)i;
    } else {
      p = (0xFFFFFFFFull << 32) | (u32)i;   // +inf-class pad, sorts last
    }
    keys[kphys(i)] = p;
  }
  __syncthreads();

  const int nchunks = NPAD / CH;
  for (int k = 2; k <= NPAD; k <<= 1) {
    // Strided LDS passes: j >= CH (pairs cross 16-element chunks).
    for (int j = k >> 1; j >= CH; j >>= 1) {
      for (int i = tid; i < NPAD; i += nt) {
        const int ixj = i ^ j;
        if (ixj > i) {
          const int pa = kphys(i), pb = kphys(ixj);
          const u64 a = keys[pa];
          const u64 b = keys[pb];
          const bool up = ((i & k) == 0);
          if (up ? (a > b) : (a < b)) { keys[pa] = b; keys[pb] = a; }
        }
      }
      __syncthreads();
    }
    // Register-fused pass: j in {8,4,2,1} clamped to k/2, within 16-chunks.
    for (int c = tid; c < nchunks; c += nt) {
      const int base = c * CH;
      u64 r[CH];
#pragma unroll
      for (int l = 0; l < CH; ++l) r[l] = keys[kphys(base + l)];
#pragma unroll
      for (int jj = 3; jj >= 0; --jj) {
        const int j = 1 << jj;
        if (j <= (k >> 1)) {
#pragma unroll
          for (int l = 0; l < CH; ++l) {
            const int lj = l ^ j;
            if (lj > l) {
              const bool up = (((base + l) & k) == 0);
              const u64 a = r[l], b = r[lj];
              if (up ? (a > b) : (a < b)) { r[l] = b; r[lj] = a; }
            }
          }
        }
      }
#pragma unroll
      for (int l = 0; l < CH; ++l) keys[kphys(base + l)] = r[l];
    }
    __syncthreads();
  }

  for (int i = tid; i < N; i += nt) {
    const u64 p   = keys[kphys(i)];
    const u32 key = (u32)(p >> 32);
    const u32 b   = (key & 0x80000000u) ? (key ^ 0x80000000u) : ~key;
    out_vals[(size_t)q * N + i] = __uint_as_float(b);
    out_idx[(size_t)q * N + i] = (int)(u32)(p & 0xFFFFFFFFu);
  }
}

// ---------------------------------------------------------------------------
extern "C" void kernel_launch(void* const* d_in, const int* in_sizes, int n_in,
                              void* d_out, int out_size, void* d_ws, size_t ws_size,
                              hipStream_t stream) {
  const float* q_feat = (const float*)d_in[0];
  const float* q_sig  = (const float*)d_in[1];
  const float* g_feat = (const float*)d_in[2];
  const float* g_sig  = (const float*)d_in[3];

  const int D = 64;
  const int Q = in_sizes[0] / D;   // 64
  const int N = in_sizes[2] / D;   // 20000

  float* neg      = (float*)d_ws;                             // [Q, N] scratch (5.12 MB)
  float* out_vals = (float*)d_out;                            // sorted -sims, ascending
  int*   out_idx  = (int*)((float*)d_out + (size_t)Q * N);    // argsort indices

  const int tiles = (N + TILE_N - 1) / TILE_N;                // 313
  const size_t lds1 = (size_t)(Q * D * 2 + TILE_N * LDS_ROW * 2) * sizeof(float); // ~64.5 KB
  elk_sims_kernel<<<tiles, 256, lds1, stream>>>(q_feat, q_sig, g_feat, g_sig, neg, N);

  int npad = 1; while (npad < N) npad <<= 1;                  // 32768
  const size_t lds2 = (size_t)(npad + (npad >> 4)) * sizeof(u64); // 272 KB dynamic LDS
  elk_sort_kernel<<<Q, 1024, lds2, stream>>>(neg, out_vals, out_idx, N, npad);
}